// MultiHeadAttention_2680059593121
// MI455X (gfx1250) — compile-verified
//
#include <hip/hip_runtime.h>
#include <math.h>

// MultiHeadAttention on gfx1250 (MI455X): fp32 end-to-end via V_WMMA_F32_16X16X4_F32,
// with double-buffered async global->LDS staging (ASYNCcnt) for GEMMs *and* K/V tiles.
// B=4, N=2048, C=1024, H=16, D=64.

#define BATCH 4
#define SEQ   2048
#define EMB   1024
#define HEADS 16
#define HD    64

typedef __attribute__((ext_vector_type(2))) float v2f;
typedef __attribute__((ext_vector_type(8))) float v8f;

__device__ __forceinline__ v8f wmma_f32_k4(v2f a, v2f b, v8f c) {
    // D = A(16x4) * B(4x16) + C(16x16), all fp32.
    return __builtin_amdgcn_wmma_f32_16x16x4_f32(
        /*neg_a=*/false, a, /*neg_b=*/false, b,
        /*c_mod=*/(short)0, c, /*reuse_a=*/false, /*reuse_b=*/false);
}

// Async copy of 16 bytes from global to LDS (per lane). Tracked by ASYNCcnt.
__device__ __forceinline__ void async_cp16(unsigned lds_off, const float* gptr) {
    asm volatile("global_load_async_to_lds_b128 %0, %1, off"
                 :: "v"(lds_off), "v"(gptr)
                 : "memory");
}

__device__ __forceinline__ unsigned lds_addr(const void* p) {
    return (unsigned)(size_t)p;   // generic LDS pointer: low 32 bits = LDS offset
}

// -----------------------------------------------------------------------------
// LDS-tiled GEMM with bias: C[M x N] = A[M x K] @ B[K x N] + bias[N]
// Block = 128 threads (4 waves) -> 64x64 C tile; wave -> 32x32 quad.
// K loop in steps of 16, double-buffered async global->LDS staging.
// -----------------------------------------------------------------------------
#define BM 64
#define BN 64
#define BK 16

__device__ __forceinline__ void stage_tiles(const float* __restrict__ A,
                                            const float* __restrict__ B,
                                            float* As, float* Bs,
                                            int m0, int n0, int k0,
                                            int K, int N, int t) {
#pragma unroll
    for (int r = 0; r < 2; ++r) {               // A tile: 64x16 f32 = 256 x 16B
        const int i = t + 128 * r;
        const int m = i >> 2;
        const int j = (i & 3) * 4;
        async_cp16(lds_addr(&As[m * BK + j]),
                   A + (size_t)(m0 + m) * K + k0 + j);
    }
#pragma unroll
    for (int r = 0; r < 2; ++r) {               // B tile: 16x64 f32 = 256 x 16B
        const int i = t + 128 * r;
        const int k = i >> 4;
        const int j = (i & 15) * 4;
        async_cp16(lds_addr(&Bs[k * BN + j]),
                   B + (size_t)(k0 + k) * N + n0 + j);
    }
}

__global__ void wmma_gemm_bias(const float* __restrict__ A,
                               const float* __restrict__ B,
                               const float* __restrict__ bias,
                               float* __restrict__ C,
                               int M, int K, int N) {
    __shared__ __align__(16) float As[2][BM * BK];   // [m][k]
    __shared__ __align__(16) float Bs[2][BK * BN];   // [k][n]

    const int t    = threadIdx.x;
    const int lane = t & 31;
    const int wave = t >> 5;
    const int lm   = lane & 15;
    const int hi   = lane >> 4;

    const int m0 = blockIdx.y * BM;
    const int n0 = blockIdx.x * BN;
    const int wm = (wave >> 1) * 32;
    const int wn = (wave & 1) * 32;

    v8f acc[2][2] = {};

    stage_tiles(A, B, As[0], Bs[0], m0, n0, 0, K, N, t);

    const int nt = K / BK;
    for (int kt = 0; kt < nt; ++kt) {
        asm volatile("s_wait_asynccnt 0" ::: "memory");
        __syncthreads();
        if (kt + 1 < nt)
            stage_tiles(A, B, As[(kt + 1) & 1], Bs[(kt + 1) & 1],
                        m0, n0, (kt + 1) * BK, K, N, t);

        const float* Ab = As[kt & 1];
        const float* Bb = Bs[kt & 1];
#pragma unroll
        for (int c = 0; c < 4; ++c) {
            const int kk = 4 * c + 2 * hi;
            v2f a0 = *(const v2f*)&Ab[(wm + lm) * BK + kk];
            v2f a1 = *(const v2f*)&Ab[(wm + 16 + lm) * BK + kk];
            v2f b0, b1;
            b0.x = Bb[kk * BN + wn + lm];
            b0.y = Bb[(kk + 1) * BN + wn + lm];
            b1.x = Bb[kk * BN + wn + 16 + lm];
            b1.y = Bb[(kk + 1) * BN + wn + 16 + lm];
            acc[0][0] = wmma_f32_k4(a0, b0, acc[0][0]);
            acc[0][1] = wmma_f32_k4(a0, b1, acc[0][1]);
            acc[1][0] = wmma_f32_k4(a1, b0, acc[1][0]);
            acc[1][1] = wmma_f32_k4(a1, b1, acc[1][1]);
        }
    }

#pragma unroll
    for (int tm = 0; tm < 2; ++tm) {
#pragma unroll
        for (int tn = 0; tn < 2; ++tn) {
            const int nb = n0 + wn + 16 * tn + lm;
            const float bv = bias[nb];
            const int mb = m0 + wm + 16 * tm + hi * 8;
#pragma unroll
            for (int r = 0; r < 8; ++r)
                C[(size_t)(mb + r) * N + nb] = acc[tm][tn][r] + bv;
        }
    }
}

// -----------------------------------------------------------------------------
// Flash attention (fp32, online softmax).
// Block = 4 waves handling 4 adjacent 16-row query blocks of the SAME (b,h);
// K/V tiles (16 keys x 64d) are shared through LDS, staged with double-buffered
// async copies -> 4x less K/V traffic and fully coalesced DMA.
// qkv layout: [B][N][3][H][HD]; out layout: [B][N][H*HD].
// -----------------------------------------------------------------------------
#define FA_WAVES 4
#define KST 68   // 64 + 4 pad floats: rows stay 16B-aligned, banks spread (4*lm+k)

__global__ void flash_attn_f32(const float* __restrict__ qkv,
                               float* __restrict__ out) {
    __shared__ __align__(16) float Ks[2][16 * KST];
    __shared__ __align__(16) float Vs[2][16 * KST];
    __shared__ float Pl[FA_WAVES][16 * 17];

    const int t    = threadIdx.x;
    const int lane = t & 31;
    const int wave = t >> 5;
    const int lm   = lane & 15;
    const int hi   = lane >> 4;

    const int nQgrp = SEQ / (16 * FA_WAVES);     // 32 q-groups per (b,h)
    const int qg = blockIdx.x % nQgrp;
    const int bh = blockIdx.x / nQgrp;
    const int h  = bh % HEADS;
    const int b  = bh / HEADS;
    const int qb = qg * FA_WAVES + wave;         // this wave's 16-row q block

    const size_t tokStride = (size_t)3 * EMB;
    const size_t headOff   = (size_t)h * HD;

    const float* Kg = qkv + (size_t)b * SEQ * tokStride + 1 * EMB + headOff;
    const float* Vg = qkv + (size_t)b * SEQ * tokStride + 2 * EMB + headOff;

    // ---- load Q (16 x 64) into A-operand registers; pre-scale by 1/sqrt(64) ----
    const float* Qbase =
        qkv + ((size_t)(b * SEQ + qb * 16 + lm)) * tokStride + 0 * EMB + headOff;
    v2f qa[16];
#pragma unroll
    for (int c = 0; c < 16; ++c) {
        v2f q = *(const v2f*)(Qbase + 4 * c + 2 * hi);
        q.x *= 0.125f;  // 1/sqrt(HD) = 1/8, exact
        q.y *= 0.125f;
        qa[c] = q;
    }

    float mstate[8], lstate[8];
#pragma unroll
    for (int r = 0; r < 8; ++r) { mstate[r] = -INFINITY; lstate[r] = 0.0f; }

    v8f o[4] = {};
    float* Plw = &Pl[wave][0];

    // Stage a 16-key K tile + V tile (each 16x64 f32 = 256 x 16B chunks; 128 thr x 2).
    auto stage_kv = [&](int buf, int kb) {
#pragma unroll
        for (int r = 0; r < 2; ++r) {
            const int i = t + 128 * r;
            const int k = i >> 4;
            const int j = (i & 15) * 4;
            async_cp16(lds_addr(&Ks[buf][k * KST + j]),
                       Kg + (size_t)(kb + k) * tokStride + j);
            async_cp16(lds_addr(&Vs[buf][k * KST + j]),
                       Vg + (size_t)(kb + k) * tokStride + j);
        }
    };

    stage_kv(0, 0);

    const int nkb = SEQ / 16;
    for (int kbi = 0; kbi < nkb; ++kbi) {
        asm volatile("s_wait_asynccnt 0" ::: "memory");  // own copies done
        __syncthreads();                                  // everyone's copies done
        if (kbi + 1 < nkb)
            stage_kv((kbi + 1) & 1, (kbi + 1) * 16);      // overlap next stage

        const float* Kb = &Ks[kbi & 1][0];
        const float* Vb = &Vs[kbi & 1][0];

        // ---- S = Q @ K^T  (16x16): B-op element (k=d, n=key) ----
        v8f s = {};
#pragma unroll
        for (int c = 0; c < 16; ++c) {
            v2f kv = *(const v2f*)&Kb[lm * KST + 4 * c + 2 * hi];  // ds_load_b64
            s = wmma_f32_k4(qa[c], kv, s);
        }

        // ---- online softmax: row reductions across 16-lane half-groups ----
#pragma unroll
        for (int r = 0; r < 8; ++r) {
            float v = s[r];
            v = fmaxf(v, __shfl_xor(v, 1, 16));
            v = fmaxf(v, __shfl_xor(v, 2, 16));
            v = fmaxf(v, __shfl_xor(v, 4, 16));
            v = fmaxf(v, __shfl_xor(v, 8, 16));
            const float mn    = fmaxf(mstate[r], v);
            const float alpha = __expf(mstate[r] - mn);
            const float p     = __expf(s[r] - mn);
            float ps = p;
            ps += __shfl_xor(ps, 1, 16);
            ps += __shfl_xor(ps, 2, 16);
            ps += __shfl_xor(ps, 4, 16);
            ps += __shfl_xor(ps, 8, 16);
            lstate[r] = lstate[r] * alpha + ps;
            mstate[r] = mn;
#pragma unroll
            for (int tt = 0; tt < 4; ++tt) o[tt][r] *= alpha;
            s[r] = p;
        }

        // ---- relayout P (C layout -> A layout) via private LDS tile ----
#pragma unroll
        for (int r = 0; r < 8; ++r) {
            Plw[(r + 8 * hi) * 17 + lm] = s[r];
        }
        asm volatile("s_wait_dscnt 0" ::: "memory");

        // ---- O += P @ V: B-op element (k=key, n=d) ----
#pragma unroll
        for (int c = 0; c < 4; ++c) {
            v2f pa;
            pa.x = Plw[lm * 17 + 4 * c + 2 * hi];
            pa.y = Plw[lm * 17 + 4 * c + 2 * hi + 1];
            const int kr = 4 * c + 2 * hi;
#pragma unroll
            for (int tt = 0; tt < 4; ++tt) {
                v2f vb;
                vb.x = Vb[kr * KST + 16 * tt + lm];
                vb.y = Vb[(kr + 1) * KST + 16 * tt + lm];
                o[tt] = wmma_f32_k4(pa, vb, o[tt]);
            }
        }
    }

    // ---- normalize and write O: out[(b,qrow)][h*64 + d] ----
    const int mBase = qb * 16 + hi * 8;
#pragma unroll
    for (int r = 0; r < 8; ++r) {
        const float inv = 1.0f / lstate[r];
        const size_t row = ((size_t)b * SEQ + mBase + r) * EMB + headOff + lm;
#pragma unroll
        for (int tt = 0; tt < 4; ++tt) {
            out[row + 16 * tt] = o[tt][r] * inv;
        }
    }
}

// -----------------------------------------------------------------------------
// Host-side launch
// -----------------------------------------------------------------------------
extern "C" void kernel_launch(void* const* d_in, const int* in_sizes, int n_in,
                              void* d_out, int out_size, void* d_ws, size_t ws_size,
                              hipStream_t stream) {
    const float* x      = (const float*)d_in[0];
    const float* w_qkv  = (const float*)d_in[1];
    const float* b_qkv  = (const float*)d_in[2];
    const float* w_proj = (const float*)d_in[3];
    const float* b_proj = (const float*)d_in[4];
    float* out = (float*)d_out;

    float* qkv  = (float*)d_ws;                                   // 96 MB
    float* attn = qkv + (size_t)BATCH * SEQ * 3 * EMB;            // +32 MB

    const int M = BATCH * SEQ;       // 8192

    // 1) qkv = x @ w_qkv + b_qkv        (8192 x 1024) @ (1024 x 3072)
    {
        dim3 grid((3 * EMB) / BN, M / BM);
        wmma_gemm_bias<<<grid, 128, 0, stream>>>(
            x, w_qkv, b_qkv, qkv, M, EMB, 3 * EMB);
    }

    // 2) attention: attn[(b,n)][h*64+d]
    {
        const int nBlocks = BATCH * HEADS * (SEQ / (16 * FA_WAVES));  // 2048
        flash_attn_f32<<<nBlocks, 32 * FA_WAVES, 0, stream>>>(qkv, attn);
    }

    // 3) out = attn @ w_proj + b_proj   (8192 x 1024) @ (1024 x 1024)
    {
        dim3 grid(EMB / BN, M / BM);
        wmma_gemm_bias<<<grid, 128, 0, stream>>>(
            attn, w_proj, b_proj, out, M, EMB, EMB);
    }
}